// MessageLayer_84018150244580
// MI455X (gfx1250) — compile-verified
//
#include <hip/hip_runtime.h>
#include <hip/hip_bf16.h>

// ext-vector types matching WMMA f32 16x16x4 VGPR footprints (wave32):
// A 16x4 f32 = 64 f32 / 32 lanes = 2 VGPRs; B 4x16 likewise; C/D 16x16 = 8 VGPRs.
typedef __attribute__((ext_vector_type(2))) float v2f;
typedef __attribute__((ext_vector_type(8))) float v8f;

#define N_ATOMS 20000
#define DIM     64

// Branchless extraction of acc[rsel] (rsel = l16 & 7) -> v_cndmask chain.
__device__ __forceinline__ float select_row(const v8f& a, int rsel) {
    float v = a[0];
#pragma unroll
    for (int r = 1; r < 8; ++r)
        v = (rsel == r) ? a[r] : v;
    return v;
}

// One wave (32 lanes) per edge. Computes messages[e] = bond[e] @ atom[send[e]]
// with V_WMMA_F32_16X16X4_F32 (4 row-tiles x 16 K-chunks), then scatter-adds
// into out[recv[e]] with full-EXEC global f32 atomics (2 per edge).
__global__ __launch_bounds__(256) void MessageLayer_wmma_kernel(
    const float* __restrict__ atom,   // [N_ATOMS][64]
    const float* __restrict__ bond,   // [n_edges][64][64]
    const int*   __restrict__ conn,   // [n_edges][2] = {recv, send}
    float*       __restrict__ out,    // [N_ATOMS][64], pre-zeroed
    int n_edges)
{
    const int wavesPerBlock = blockDim.x >> 5;
    const int e = blockIdx.x * wavesPerBlock + (threadIdx.x >> 5);
    if (e >= n_edges) return;   // wave-uniform: EXEC stays all-ones past here

    const int lane = threadIdx.x & 31;
    const int half = lane >> 4;   // 0: K pair {0,1} / rows M..M+7 ; 1: K pair {2,3} / rows +8
    const int l16  = lane & 15;
    const int rsel = l16 & 7;

    const int2 rc  = ((const int2*)conn)[e];   // {recv, send}
    const int recv = rc.x;
    const int send = rc.y;

    const float* __restrict__ v = atom + (long)send * DIM;
    const float* __restrict__ M = bond + (long)e * (DIM * DIM);
    float* __restrict__ outRow  = out  + (long)recv * DIM;

    // Preload the gathered vector into B-operand registers, broadcast over N.
    // B (4x16, KxN) layout: VGPR0 = K=2*half, VGPR1 = K=2*half+1, any lane/column.
    v2f bv[16];
#pragma unroll
    for (int c = 0; c < 16; ++c)
        bv[c] = *(const v2f*)(v + 4 * c + 2 * half);

#pragma unroll
    for (int t = 0; t < 4; t += 2) {
        // A (16x4) layout: lane (l16,half) holds A[M=l16][K=2*half], A[.][K=2*half+1]
        const float* __restrict__ rowp0 = M + (16 * (t + 0) + l16) * DIM + 2 * half;
        const float* __restrict__ rowp1 = M + (16 * (t + 1) + l16) * DIM + 2 * half;

        v8f acc0 = {0.f, 0.f, 0.f, 0.f, 0.f, 0.f, 0.f, 0.f};
        v8f acc1 = {0.f, 0.f, 0.f, 0.f, 0.f, 0.f, 0.f, 0.f};
#pragma unroll
        for (int c = 0; c < 16; ++c) {
            v2f a0 = *(const v2f*)(rowp0 + 4 * c);
            // 8 args: (neg_a, A, neg_b, B, c_mod, C, reuse_a, reuse_b)
            acc0 = __builtin_amdgcn_wmma_f32_16x16x4_f32(
                false, a0, false, bv[c], (short)0, acc0, false, false);
        }
#pragma unroll
        for (int c = 0; c < 16; ++c) {
            v2f a1 = *(const v2f*)(rowp1 + 4 * c);
            acc1 = __builtin_amdgcn_wmma_f32_16x16x4_f32(
                false, a1, false, bv[c], (short)0, acc1, false, false);
        }

        // D layout: VGPR r, lanes of half h -> row 16t + r + 8h, replicated over
        // the 16 lanes of each half (all B columns identical). Merge both tiles:
        // lanes l16<8 carry tile t, lanes l16>=8 carry tile t+1 -> 32 distinct
        // rows per atomic, full EXEC, no exec-mask juggling.
        const float vt0 = select_row(acc0, rsel);
        const float vt1 = select_row(acc1, rsel);
        const float val = (l16 < 8) ? vt0 : vt1;
        const int   row = 16 * t + ((l16 >> 3) << 4) + rsel + 8 * half;
        atomicAdd(outRow + row, val);
    }
}

extern "C" void kernel_launch(void* const* d_in, const int* in_sizes, int n_in,
                              void* d_out, int out_size, void* d_ws, size_t ws_size,
                              hipStream_t stream)
{
    const float* atom = (const float*)d_in[0];   // 20000*64 f32
    const float* bond = (const float*)d_in[1];   // 100000*64*64 f32
    const int*   conn = (const int*)d_in[2];     // 100000*2 i32
    float*       out  = (float*)d_out;           // 20000*64 f32

    const int n_edges = in_sizes[2] / 2;

    // Atomics accumulate -> output must start at zero every call.
    hipMemsetAsync(out, 0, (size_t)out_size * sizeof(float), stream);

    const int threads = 256;                     // 8 waves/block (wave32)
    const int wavesPerBlock = threads / 32;
    const int grid = (n_edges + wavesPerBlock - 1) / wavesPerBlock;  // 12500

    MessageLayer_wmma_kernel<<<grid, threads, 0, stream>>>(atom, bond, conn, out, n_edges);
}